// Model_28406913696306
// MI455X (gfx1250) — compile-verified
//
#include <hip/hip_runtime.h>
#include <math.h>

// Problem constants (from reference)
#define V_  128001
#define D_  128
#define H_  128
#define T_  64            // scan steps (reference "B")
#define R_  200           // rows per step (reference "S", the carried batch)
#define N_  (T_ * R_)     // 12800 nodes
#define E_  (N_ * 32)     // 409600 edges

#define HS  132           // LDS row stride for 128-wide state tiles (bank spread: 132%64=4)
#define GS  516           // LDS row stride for 512-wide gate tiles (8*516%64=32)

typedef float v2f __attribute__((ext_vector_type(2)));
typedef float v8f __attribute__((ext_vector_type(8)));

// fp32 WMMA: D(16x16) = A(16x4) * B(4x16) + C  -- exact-precision matrix path on CDNA5
__device__ __forceinline__ v8f wmma4(v2f a, v2f b, v8f c) {
    return __builtin_amdgcn_wmma_f32_16x16x4_f32(
        /*neg_a=*/false, a, /*neg_b=*/false, b,
        /*c_mod=*/(short)0, c, /*reuse_a=*/false, /*reuse_b=*/false);
}

__device__ __forceinline__ float sigmoid_(float x) { return 1.0f / (1.0f + __expf(-x)); }

// ---------------- Stage 1: degrees via float atomics ----------------
__global__ void k_degrees(const int* __restrict__ src, const int* __restrict__ dst,
                          float* __restrict__ deg_out, float* __restrict__ deg_in) {
    int e = blockIdx.x * blockDim.x + threadIdx.x;
    if (e < E_) {
        atomicAdd(&deg_out[src[e]], 1.0f);
        atomicAdd(&deg_in[dst[e]], 1.0f);
    }
}

__global__ void k_norms(const float* __restrict__ deg, float* __restrict__ nrm) {
    int i = blockIdx.x * blockDim.x + threadIdx.x;
    if (i < N_) {
        float d = deg[i];
        nrm[i] = (d > 0.0f) ? rsqrtf(fmaxf(d, 1.0f)) : 0.0f;
    }
}

// ---------------- Stage 2: embedding gather * n_src ----------------
__global__ void k_gather(const int* __restrict__ nodes, const float* __restrict__ emb,
                         const float* __restrict__ nsrc, float* __restrict__ xs) {
    int gid = blockIdx.x * blockDim.x + threadIdx.x; // N_*32 threads, float4 each
    if (gid < N_ * 32) {
        int n = gid >> 5, c = gid & 31;
        const float4* er = (const float4*)(emb + (long)nodes[n] * D_);
        float4 v = er[c];
        float s = nsrc[n];
        float4 o = make_float4(v.x * s, v.y * s, v.z * s, v.w * s);
        ((float4*)(xs + (long)n * D_))[c] = o;
    }
}

// ---------------- Stage 3: scatter-add over edges (L2 atomics) ----------------
__global__ void k_scatter(const int* __restrict__ src, const int* __restrict__ dst,
                          const float* __restrict__ xs, float* __restrict__ agg) {
    long gid = (long)blockIdx.x * blockDim.x + threadIdx.x;
    if (gid < (long)E_ * 32) {
        int e = (int)(gid >> 5), c = (int)(gid & 31);
        int s = src[e], d = dst[e];
        float4 v = ((const float4*)(xs + (long)s * D_))[c];
        float* base = agg + (long)d * D_ + c * 4;
        atomicAdd(base + 0, v.x);
        atomicAdd(base + 1, v.y);
        atomicAdd(base + 2, v.z);
        atomicAdd(base + 3, v.w);
    }
}

// ---------------- Stage 4: h = relu((agg * n_dst) @ gW + gb), WMMA fp32 ----------------
// One wave per 16x16 output tile; tiles = (N_/16) x (D_/16) = 800 x 8 = 6400 waves.
__global__ void k_gcn_gemm(const float* __restrict__ agg, const float* __restrict__ ndst,
                           const float* __restrict__ gW, const float* __restrict__ gb,
                           float* __restrict__ h) {
    int wave = blockIdx.x * (blockDim.x >> 5) + (threadIdx.x >> 5);
    int lane = threadIdx.x & 31;
    if (wave >= (N_ / 16) * (D_ / 16)) return; // wave-uniform
    int tm = wave >> 3, tn = wave & 7;
    int m0 = tm * 16, n0 = tn * 16;
    int half = lane >> 4, mr = lane & 15;
    int m = m0 + mr;
    float scale = ndst[m];
    v8f acc = {0.f, 0.f, 0.f, 0.f, 0.f, 0.f, 0.f, 0.f};
    for (int k0 = 0; k0 < D_; k0 += 4) {
        v2f a = *(const v2f*)(agg + (long)m * D_ + k0 + 2 * half);
        a.x *= scale; a.y *= scale;
        v2f b;
        b.x = gW[(k0 + 2 * half) * D_ + n0 + mr];
        b.y = gW[(k0 + 2 * half + 1) * D_ + n0 + mr];
        acc = wmma4(a, b, acc);
    }
    int col = n0 + mr;
    float bias = gb[col];
#pragma unroll
    for (int r = 0; r < 8; r++) {
        int mm = m0 + r + half * 8;
        float v = acc[r] + bias;
        h[(long)mm * D_ + col] = v > 0.f ? v : 0.f;
    }
}

// ---------------- Stage 5: persistent fused 2-layer LSTM scan (ONE launch) ----------------
// The recurrence is row-local (row m of h_t depends only on row m of h_{t-1}), so each block
// owns a 16-row tile and runs all 64 steps x 2 layers privately. Carry state lives in LDS for
// the whole scan; the 16x512 gate buffer is double-buffered so each phase needs 2 barriers.
// 13 blocks x 512 threads (16 waves); wave w computes gate columns [32w, 32w+31].
__global__ void __launch_bounds__(512) k_lstm_all(
    const float* __restrict__ hgr,
    const float* __restrict__ Wih0, const float* __restrict__ Whh0,
    const float* __restrict__ bih0, const float* __restrict__ bhh0,
    const float* __restrict__ Wih1, const float* __restrict__ Whh1,
    const float* __restrict__ bih1, const float* __restrict__ bhh1,
    const float* __restrict__ h0g, const float* __restrict__ c0g,
    float* __restrict__ olast) {
    extern __shared__ float lds[];
    float* hA  = lds;               // 16*HS  layer-0 h
    float* cA  = hA + 16 * HS;      // 16*HS  layer-0 c
    float* hB  = cA + 16 * HS;      // 16*HS  layer-1 h
    float* cB  = hB + 16 * HS;      // 16*HS  layer-1 c
    float* g0  = cB + 16 * HS;      // 16*GS  gate buffer (layer-0 phases)
    float* g1  = g0 + 16 * GS;      // 16*GS  gate buffer (layer-1 phases)
    float* bs0 = g1 + 16 * GS;      // 512    bih0+bhh0
    float* bs1 = bs0 + 512;         // 512    bih1+bhh1

    const int tid  = threadIdx.x;
    const int widx = tid >> 5, lane = tid & 31;
    const int half = lane >> 4, mr = lane & 15;
    const int m0 = blockIdx.x * 16;
    const int koff = 2 * half;

    for (int i = tid; i < 512; i += blockDim.x) {
        bs0[i] = bih0[i] + bhh0[i];
        bs1[i] = bih1[i] + bhh1[i];
    }
    for (int i = tid; i < 16 * H_; i += blockDim.x) {
        int r = i >> 7, j = i & 127;
        int mg = m0 + r;
        float a = 0.f, b = 0.f, c = 0.f, d = 0.f;
        if (mg < R_) {
            a = h0g[(long)mg * H_ + j];
            b = c0g[(long)mg * H_ + j];
            c = h0g[(long)(R_ + mg) * H_ + j];   // layer-1 slice of (L,S,H)
            d = c0g[(long)(R_ + mg) * H_ + j];
        }
        hA[r * HS + j] = a; cA[r * HS + j] = b;
        hB[r * HS + j] = c; cB[r * HS + j] = d;
    }
    __syncthreads();

    const int  m      = m0 + mr;
    const bool mvalid = (m < R_);
    const int  n0a    = (widx * 2) * 16 + mr;   // first owned gate column tile
    const int  n0b    = n0a + 16;               // second owned gate column tile
    const v8f  vz     = {0.f, 0.f, 0.f, 0.f, 0.f, 0.f, 0.f, 0.f};

    for (int t = 0; t < T_; ++t) {
        // ================= layer 0 =================
        {
            const float* xt = hgr + ((long)t * R_ + m) * D_;
            v8f acc0 = vz, acc1 = vz;
            for (int k0 = 0; k0 < H_; k0 += 4) {
                v2f ax = {0.f, 0.f};
                if (mvalid) ax = *(const v2f*)(xt + k0 + koff);
                v2f ah  = *(const v2f*)&hA[mr * HS + k0 + koff];
                v2f bia = *(const v2f*)(Wih0 + (long)n0a * D_ + k0 + koff);
                v2f bha = *(const v2f*)(Whh0 + (long)n0a * H_ + k0 + koff);
                v2f bib = *(const v2f*)(Wih0 + (long)n0b * D_ + k0 + koff);
                v2f bhb = *(const v2f*)(Whh0 + (long)n0b * H_ + k0 + koff);
                acc0 = wmma4(ax, bia, acc0);
                acc0 = wmma4(ah, bha, acc0);
                acc1 = wmma4(ax, bib, acc1);
                acc1 = wmma4(ah, bhb, acc1);
            }
#pragma unroll
            for (int r = 0; r < 8; r++) {
                g0[(r + half * 8) * GS + n0a] = acc0[r];
                g0[(r + half * 8) * GS + n0b] = acc1[r];
            }
            __syncthreads();
            for (int i = tid; i < 16 * H_; i += blockDim.x) {
                int ml = i >> 7, j = i & 127;
                if (m0 + ml < R_) {
                    const float* gr = g0 + ml * GS;
                    float gi = gr[j]           + bs0[j];
                    float gf = gr[H_ + j]      + bs0[H_ + j];
                    float gg = gr[2 * H_ + j]  + bs0[2 * H_ + j];
                    float go = gr[3 * H_ + j]  + bs0[3 * H_ + j];
                    float cn = sigmoid_(gf) * cA[ml * HS + j] + sigmoid_(gi) * tanhf(gg);
                    float hn = sigmoid_(go) * tanhf(cn);
                    cA[ml * HS + j] = cn;
                    hA[ml * HS + j] = hn;
                }
            }
            __syncthreads();
        }
        // ================= layer 1 =================
        {
            v8f acc0 = vz, acc1 = vz;
            for (int k0 = 0; k0 < H_; k0 += 4) {
                v2f ax  = *(const v2f*)&hA[mr * HS + k0 + koff];   // xt = layer-0 h (fresh)
                v2f ah  = *(const v2f*)&hB[mr * HS + k0 + koff];
                v2f bia = *(const v2f*)(Wih1 + (long)n0a * D_ + k0 + koff);
                v2f bha = *(const v2f*)(Whh1 + (long)n0a * H_ + k0 + koff);
                v2f bib = *(const v2f*)(Wih1 + (long)n0b * D_ + k0 + koff);
                v2f bhb = *(const v2f*)(Whh1 + (long)n0b * H_ + k0 + koff);
                acc0 = wmma4(ax, bia, acc0);
                acc0 = wmma4(ah, bha, acc0);
                acc1 = wmma4(ax, bib, acc1);
                acc1 = wmma4(ah, bhb, acc1);
            }
#pragma unroll
            for (int r = 0; r < 8; r++) {
                g1[(r + half * 8) * GS + n0a] = acc0[r];
                g1[(r + half * 8) * GS + n0b] = acc1[r];
            }
            __syncthreads();
            for (int i = tid; i < 16 * H_; i += blockDim.x) {
                int ml = i >> 7, j = i & 127;
                int mg = m0 + ml;
                if (mg < R_) {
                    const float* gr = g1 + ml * GS;
                    float gi = gr[j]           + bs1[j];
                    float gf = gr[H_ + j]      + bs1[H_ + j];
                    float gg = gr[2 * H_ + j]  + bs1[2 * H_ + j];
                    float go = gr[3 * H_ + j]  + bs1[3 * H_ + j];
                    float cn = sigmoid_(gf) * cB[ml * HS + j] + sigmoid_(gi) * tanhf(gg);
                    float hn = sigmoid_(go) * tanhf(cn);
                    cB[ml * HS + j] = cn;
                    hB[ml * HS + j] = hn;
                    if (mg == R_ - 1) olast[(long)t * H_ + j] = hn;  // ys1[:, -1, :]
                }
            }
            __syncthreads();
        }
    }
}

// ---------------- Stage 6: logits = out_last(64x128) @ fcW^T + fcb ----------------
// A is staged in LDS once per block (kills ~256 MB of redundant L2 A-reads); fcW's single
// 65.5 MB stream is the only large HBM read (~2.8 us floor at 23.3 TB/s). One wave per
// 16-column tile computes all four 16-row M-tiles.
__global__ void k_fc(const float* __restrict__ A, const float* __restrict__ fcW,
                     const float* __restrict__ fcb, float* __restrict__ out) {
    __shared__ float As[64 * HS];
    for (int i = threadIdx.x; i < 64 * H_; i += blockDim.x)
        As[(i >> 7) * HS + (i & 127)] = A[i];
    __syncthreads();

    int wave = blockIdx.x * (blockDim.x >> 5) + (threadIdx.x >> 5);
    int lane = threadIdx.x & 31;
    if (wave >= (V_ + 15) / 16) return; // wave-uniform
    int half = lane >> 4, mr = lane & 15;
    int n = wave * 16 + mr;
    bool nvalid = (n < V_);

    v8f acc[4];
#pragma unroll
    for (int t = 0; t < 4; t++) acc[t] = (v8f){0.f, 0.f, 0.f, 0.f, 0.f, 0.f, 0.f, 0.f};

    for (int k0 = 0; k0 < H_; k0 += 4) {
        v2f b = {0.f, 0.f};
        if (nvalid) b = *(const v2f*)(fcW + (long)n * H_ + k0 + 2 * half); // B[k][n]=fcW[n][k]
#pragma unroll
        for (int mt = 0; mt < 4; mt++) {
            v2f a = *(const v2f*)&As[(mt * 16 + mr) * HS + k0 + 2 * half];
            acc[mt] = wmma4(a, b, acc[mt]);
        }
    }
    if (!nvalid) return; // per-lane guard only after all WMMAs (EXEC full during WMMA)
    float bias = fcb[n];
#pragma unroll
    for (int mt = 0; mt < 4; mt++)
#pragma unroll
        for (int r = 0; r < 8; r++)
            out[(long)(mt * 16 + r + half * 8) * V_ + n] = acc[mt][r] + bias;
}

extern "C" void kernel_launch(void* const* d_in, const int* in_sizes, int n_in,
                              void* d_out, int out_size, void* d_ws, size_t ws_size,
                              hipStream_t stream) {
    const int*   graph_nodes = (const int*)d_in[0];
    const int*   src  = (const int*)d_in[1];
    const int*   dst  = (const int*)d_in[2];
    const float* emb  = (const float*)d_in[3];
    const float* gW   = (const float*)d_in[4];
    const float* gb   = (const float*)d_in[5];
    const float* Wih0 = (const float*)d_in[6];
    const float* Whh0 = (const float*)d_in[7];
    const float* bih0 = (const float*)d_in[8];
    const float* bhh0 = (const float*)d_in[9];
    const float* Wih1 = (const float*)d_in[10];
    const float* Whh1 = (const float*)d_in[11];
    const float* bih1 = (const float*)d_in[12];
    const float* bhh1 = (const float*)d_in[13];
    const float* fcW  = (const float*)d_in[14];
    const float* fcb  = (const float*)d_in[15];
    const float* h0   = (const float*)d_in[16];
    const float* c0   = (const float*)d_in[17];
    float* out = (float*)d_out;

    // Workspace layout (~20 MB total)
    float* p = (float*)d_ws;
    float* deg_out = p; p += N_;
    float* deg_in  = p; p += N_;          // contiguous with deg_out (single memset)
    float* n_src   = p; p += N_;
    float* n_dst   = p; p += N_;
    float* xs      = p; p += (size_t)N_ * D_;
    float* agg     = p; p += (size_t)N_ * D_;
    float* hgr     = p; p += (size_t)N_ * D_;
    float* olast   = p; p += (size_t)T_ * H_;      // (64,128)

    hipMemsetAsync(deg_out, 0, 2 * (size_t)N_ * sizeof(float), stream);
    hipMemsetAsync(agg, 0, (size_t)N_ * D_ * sizeof(float), stream);

    k_degrees<<<(E_ + 255) / 256, 256, 0, stream>>>(src, dst, deg_out, deg_in);
    k_norms  <<<(N_ + 255) / 256, 256, 0, stream>>>(deg_out, n_src);
    k_norms  <<<(N_ + 255) / 256, 256, 0, stream>>>(deg_in, n_dst);
    k_gather <<<(N_ * 32 + 255) / 256, 256, 0, stream>>>(graph_nodes, emb, n_src, xs);
    k_scatter<<<(int)(((long)E_ * 32 + 255) / 256), 256, 0, stream>>>(src, dst, xs, agg);
    k_gcn_gemm<<<800, 256, 0, stream>>>(agg, n_dst, gW, gb, hgr);

    // Entire 64-step, 2-layer LSTM scan in ONE launch (row-local recurrence).
    size_t lstm_lds = (size_t)(4 * 16 * HS + 2 * 16 * GS + 1024) * sizeof(float); // ~102 KB
    k_lstm_all<<<13, 512, lstm_lds, stream>>>(hgr,
                                              Wih0, Whh0, bih0, bhh0,
                                              Wih1, Whh1, bih1, bhh1,
                                              h0, c0, olast);

    k_fc<<<1001, 256, 0, stream>>>(olast, fcW, fcb, out);
}